// VectorQuantizer_51049981281395
// MI455X (gfx1250) — compile-verified
//
#include <hip/hip_runtime.h>

// CDNA5 / gfx1250, wave32.
typedef __attribute__((ext_vector_type(2))) float v2f;
typedef __attribute__((ext_vector_type(8))) float v8f;

#define EMB_D 64          // embedding dim
#define KT    16          // codes per WMMA tile (N)
#define ROWS_PER_WAVE 32  // two 16-row WMMA tiles per wave
#define WAVES_PER_BLOCK 4

// ---------------------------------------------------------------------------
// Kernel 1: codebook squared norms  ||c_k||^2  -> d_ws
// ---------------------------------------------------------------------------
__global__ void __launch_bounds__(256)
vq_cbnorm_kernel(const float* __restrict__ cb, float* __restrict__ norms, int K) {
    int k = blockIdx.x * blockDim.x + threadIdx.x;
    if (k < K) {
        const v2f* row = (const v2f*)(cb + (size_t)k * EMB_D);
        float s = 0.f;
#pragma unroll
        for (int i = 0; i < EMB_D / 2; ++i) {
            v2f v = row[i];
            s += v.x * v.x + v.y * v.y;
        }
        norms[k] = s;
    }
}

// ---------------------------------------------------------------------------
// Kernel 2: fused distance-GEMM (fp32 WMMA) + argmin + gather + loss
//
// A fragment (16x4 fp32): lane L holds row (L&15); lanes 0-15 carry K={4t,4t+1},
// lanes 16-31 carry K={4t+2,4t+3}  -> one v2f per K-tile t.
// B fragment (4x16 fp32): lane L holds column (L&15) == codebook row; same
// K-pair-per-lane-half layout -> loads look identical to A but over codebook.
// C/D (16x16 fp32, v8f): VGPR v: lanes 0-15 -> M=v, lanes 16-31 -> M=v+8; N=L&15.
// ---------------------------------------------------------------------------
__global__ void __launch_bounds__(WAVES_PER_BLOCK * 32)
vq_argmin_kernel(const float* __restrict__ z,
                 const float* __restrict__ cb,
                 const float* __restrict__ cbnorm,
                 float* __restrict__ zq_out,     // [N, 64]
                 float* __restrict__ loss_out,   // [1] (pre-zeroed)
                 float* __restrict__ idx_out,    // [N] indices as float
                 int K, float loss_scale) {
    const int lane  = threadIdx.x & 31;
    const int wave  = threadIdx.x >> 5;
    const int lrow  = lane & 15;          // row within tile / column within tile
    const int khalf = (lane >> 4) << 1;   // 0 or 2: which K pair this lane holds

    const size_t tile0 = ((size_t)blockIdx.x * WAVES_PER_BLOCK + wave) * ROWS_PER_WAVE;
    const size_t tile1 = tile0 + 16;

    // ---- load A fragments for two 16-row tiles (resident in VGPRs) ----
    v2f a0[16], a1[16];
#pragma unroll
    for (int t = 0; t < 16; ++t) {
        a0[t] = *(const v2f*)(z + (tile0 + lrow) * EMB_D + 4 * t + khalf);
        a1[t] = *(const v2f*)(z + (tile1 + lrow) * EMB_D + 4 * t + khalf);
    }

    float bs0[8], bs1[8];
    int   bi0[8], bi1[8];
#pragma unroll
    for (int v = 0; v < 8; ++v) {
        bs0[v] = 3.0e38f; bs1[v] = 3.0e38f;
        bi0[v] = 0;       bi1[v] = 0;
    }

    // ---- sweep codebook in 16-column tiles ----
    const int ntiles = K / KT;
    for (int j = 0; j < ntiles; ++j) {
        const int code = j * KT + lrow;   // this lane's codebook row (= B column)

        v2f b[16];
#pragma unroll
        for (int t = 0; t < 16; ++t)
            b[t] = *(const v2f*)(cb + (size_t)code * EMB_D + 4 * t + khalf);

        v8f acc0 = {0.f, 0.f, 0.f, 0.f, 0.f, 0.f, 0.f, 0.f};
        v8f acc1 = {0.f, 0.f, 0.f, 0.f, 0.f, 0.f, 0.f, 0.f};
#pragma unroll
        for (int t = 0; t < 16; ++t) {
            acc0 = __builtin_amdgcn_wmma_f32_16x16x4_f32(
                false, a0[t], false, b[t], (short)0, acc0, false, false);
            acc1 = __builtin_amdgcn_wmma_f32_16x16x4_f32(
                false, a1[t], false, b[t], (short)0, acc1, false, false);
        }

        const float cn = cbnorm[code];    // ||c||^2 for this lane's column
#pragma unroll
        for (int v = 0; v < 8; ++v) {
            float s0 = cn - 2.0f * acc0[v];
            float s1 = cn - 2.0f * acc1[v];
            if (s0 < bs0[v]) { bs0[v] = s0; bi0[v] = code; }
            if (s1 < bs1[v]) { bs1[v] = s1; bi1[v] = code; }
        }
    }

    // ---- cross-lane argmin over the 16 columns (butterfly within each half) ----
#pragma unroll
    for (int v = 0; v < 8; ++v) {
#pragma unroll
        for (int off = 8; off >= 1; off >>= 1) {
            float os0 = __shfl_xor(bs0[v], off, 32);
            int   oi0 = __shfl_xor(bi0[v], off, 32);
            if (os0 < bs0[v] || (os0 == bs0[v] && oi0 < bi0[v])) { bs0[v] = os0; bi0[v] = oi0; }
            float os1 = __shfl_xor(bs1[v], off, 32);
            int   oi1 = __shfl_xor(bi1[v], off, 32);
            if (os1 < bs1[v] || (os1 == bs1[v] && oi1 < bi1[v])) { bs1[v] = os1; bi1[v] = oi1; }
        }
    }
    // Lanes 0-15 now hold winners for rows 0..7 (index v); lanes 16-31 for rows 8..15.

    // ---- gather winning codebook rows, write z_q (== z_q_st), loss partial ----
    float lsum = 0.f;
#pragma unroll
    for (int r = 0; r < 16; ++r) {
        const int src  = (r < 8) ? 0 : 16;
        // tile 0
        {
            int    kidx = __shfl(bi0[r & 7], src, 32);
            size_t row  = tile0 + r;
            v2f cv = *(const v2f*)(cb + (size_t)kidx * EMB_D + lane * 2);
            v2f zv = *(const v2f*)(z  + row * EMB_D + lane * 2);
            *(v2f*)(zq_out + row * EMB_D + lane * 2) = cv;
            float dx = cv.x - zv.x, dy = cv.y - zv.y;
            lsum += dx * dx + dy * dy;
            if (lane == 0) idx_out[row] = (float)kidx;
        }
        // tile 1
        {
            int    kidx = __shfl(bi1[r & 7], src, 32);
            size_t row  = tile1 + r;
            v2f cv = *(const v2f*)(cb + (size_t)kidx * EMB_D + lane * 2);
            v2f zv = *(const v2f*)(z  + row * EMB_D + lane * 2);
            *(v2f*)(zq_out + row * EMB_D + lane * 2) = cv;
            float dx = cv.x - zv.x, dy = cv.y - zv.y;
            lsum += dx * dx + dy * dy;
            if (lane == 0) idx_out[row] = (float)kidx;
        }
    }

    // ---- loss: wave reduce + one atomic per wave ----
#pragma unroll
    for (int off = 16; off >= 1; off >>= 1)
        lsum += __shfl_xor(lsum, off, 32);
    if (lane == 0)
        atomicAdd(loss_out, lsum * loss_scale);
}

// ---------------------------------------------------------------------------
extern "C" void kernel_launch(void* const* d_in, const int* in_sizes, int n_in,
                              void* d_out, int out_size, void* d_ws, size_t ws_size,
                              hipStream_t stream) {
    (void)n_in; (void)out_size; (void)ws_size;

    const float* z  = (const float*)d_in[0];   // [N, 64]
    const float* cb = (const float*)d_in[1];   // [K, 64]
    const int ND = in_sizes[0];                // N * 64
    const int KD = in_sizes[1];                // K * 64
    const int N  = ND / EMB_D;
    const int K  = KD / EMB_D;

    float* zq   = (float*)d_out;               // [N*64] z_q_st
    float* loss = zq + (size_t)ND;             // [1]
    float* idxo = loss + 1;                    // [N] indices (as float)
    float* cbn  = (float*)d_ws;                // [K] codebook norms

    // loss = (1 + beta) * mean((z_q - z)^2)
    const float loss_scale = 1.25f / (float)ND;

    hipMemsetAsync(loss, 0, sizeof(float), stream);

    vq_cbnorm_kernel<<<(K + 255) / 256, 256, 0, stream>>>(cb, cbn, K);

    const int rows_per_block = WAVES_PER_BLOCK * ROWS_PER_WAVE;  // 128
    vq_argmin_kernel<<<N / rows_per_block, WAVES_PER_BLOCK * 32, 0, stream>>>(
        z, cb, cbn, zq, loss, idxo, K, loss_scale);
}